// TrilinearMatrixAttention_12266426597383
// MI455X (gfx1250) — compile-verified
//
#include <hip/hip_runtime.h>

typedef __attribute__((ext_vector_type(2))) float v2f;
typedef __attribute__((ext_vector_type(8))) float v8f;

#define Bsz 2
#define Ssz 384
#define Dsz 256
#define NT  (Ssz / 16)          // 24 tiles per dim
#define NROWS (Bsz * Ssz)       // 768 rows per matrix
#define NPART 2048              // W-reduction partials

// ---------------- Stage 1a: partial sums of W (deterministic fixed order) ----
__global__ void reduce_w_partials(const float* __restrict__ W,
                                  float* __restrict__ partials, int n) {
    __shared__ float sm[256];
    const int tid = threadIdx.x;
    const size_t gid = (size_t)blockIdx.x * blockDim.x + tid;
    const size_t stride = (size_t)gridDim.x * blockDim.x;
    const float4* W4 = (const float4*)W;
    const size_t n4 = (size_t)n >> 2;   // n = 256^3, divisible by 4
    float acc = 0.f;
    for (size_t i = gid; i < n4; i += stride) {
        float4 v = W4[i];
        acc += v.x; acc += v.y; acc += v.z; acc += v.w;   // fixed order
    }
    sm[tid] = acc;
    __syncthreads();
    for (int off = 128; off > 0; off >>= 1) {
        if (tid < off) sm[tid] += sm[tid + off];
        __syncthreads();
    }
    if (tid == 0) partials[blockIdx.x] = sm[0];
}

// ---------------- Stage 1b: finalize wsum (single block, deterministic) -----
__global__ void reduce_w_final(const float* __restrict__ partials,
                               float* __restrict__ wsum, int nparts) {
    __shared__ float sm[256];
    const int tid = threadIdx.x;
    float acc = 0.f;
    for (int i = tid; i < nparts; i += 256) acc += partials[i];
    sm[tid] = acc;
    __syncthreads();
    for (int off = 128; off > 0; off >>= 1) {
        if (tid < off) sm[tid] += sm[tid + off];
        __syncthreads();
    }
    if (tid == 0) wsum[0] = sm[0];
}

// ---------------- Stage 2: row sums s1,s2,s3 (one wave32 per row) -----------
__global__ void row_sums(const float* __restrict__ m1,
                         const float* __restrict__ m2,
                         const float* __restrict__ m3,
                         float* __restrict__ ssums) {
    const int wave = (int)((blockIdx.x * blockDim.x + threadIdx.x) >> 5);
    const int lane = threadIdx.x & 31;
    if (wave >= 3 * NROWS) return;
    const float* src = (wave < NROWS) ? m1 : (wave < 2 * NROWS ? m2 : m3);
    const int row = wave % NROWS;
    const float* p = src + (size_t)row * Dsz;
    float acc = 0.f;
#pragma unroll
    for (int k = 0; k < Dsz; k += 32) acc += p[k + lane];
    // wave32 butterfly reduction (fixed order -> deterministic)
#pragma unroll
    for (int off = 16; off > 0; off >>= 1) acc += __shfl_xor(acc, off, 32);
    if (lane == 0) ssums[wave] = acc;
}

// ---------------- Stage 3: rank-1 masked write, 453 MB of NT stores ---------
// One block per (b, jt, kt) 16x16 tile; the s2(j) x s3(k) outer product is
// formed once per wave with V_WMMA_F32_16X16X4_F32 (only K=0 populated),
// masked with k>j, then scaled by coef*s1[b,i] for the wave's i-chunk.
__global__ void trilinear_write(const float* __restrict__ ssums,
                                const float* __restrict__ wsum_p,
                                float* __restrict__ out) {
    const int tile = blockIdx.x;
    const int b    = tile / (NT * NT);
    const int rem  = tile - b * NT * NT;
    const int jt   = rem / NT;
    const int kt   = rem - jt * NT;

    const int lane   = threadIdx.x & 31;
    const int waveId = threadIdx.x >> 5;
    const int half   = lane >> 4;     // 0: lanes 0-15, 1: lanes 16-31
    const int l16    = lane & 15;

    const float* s1 = ssums;
    const float* s2 = ssums + NROWS;
    const float* s3 = ssums + 2 * NROWS;

    const int jbase = jt * 16;
    const int kbase = kt * 16;

    // A(16x4): rows M=j hold s2 in column K=0 (VGPR0, lanes 0-15); rest zero.
    // B(4x16): row K=0 holds s3 across N=k (VGPR0, lanes 0-15); rest zero.
    const float aval = s2[b * Ssz + jbase + l16];
    const float bval = s3[b * Ssz + kbase + l16];
    v2f a;  a.x  = half ? 0.f : aval;  a.y  = 0.f;
    v2f bm; bm.x = half ? 0.f : bval;  bm.y = 0.f;
    v8f c = {};
    v8f d = __builtin_amdgcn_wmma_f32_16x16x4_f32(
        /*neg_a=*/false, a, /*neg_b=*/false, bm,
        /*c_mod=*/(short)0, c, /*reuse_a=*/false, /*reuse_b=*/false);

    // C/D layout: lane -> N(k)=lane%16 ; VGPR v -> M(j)=v+8*(lane>=16).
    const int kg = kbase + l16;
    float  tilev[8];
    size_t offs[8];
#pragma unroll
    for (int v = 0; v < 8; ++v) {
        const int jg = jbase + v + 8 * half;
        tilev[v] = (kg > jg) ? d[v] : 0.f;              // triu(k>j) mask
        offs[v]  = (size_t)jg * Ssz + kg;
    }

    const float coef = wsum_p[0] * (1.0f / 16.0f);      // wsum / sqrt(256)

    const int ichunk = Ssz / 8;                          // 48 i per wave
    const int i0 = waveId * ichunk;
    const size_t plane = (size_t)Ssz * Ssz;              // 147456
    float* p = out + ((size_t)b * Ssz + i0) * plane;
    for (int ii = 0; ii < ichunk; ++ii) {
        const float ci = coef * s1[b * Ssz + i0 + ii];   // uniform -> scalar load
#pragma unroll
        for (int v = 0; v < 8; ++v) {
            __builtin_nontemporal_store(ci * tilev[v], p + offs[v]);
        }
        p += plane;
    }
}

// ---------------- Launch ----------------------------------------------------
extern "C" void kernel_launch(void* const* d_in, const int* in_sizes, int n_in,
                              void* d_out, int out_size, void* d_ws, size_t ws_size,
                              hipStream_t stream) {
    const float* m1 = (const float*)d_in[0];
    const float* m2 = (const float*)d_in[1];
    const float* m3 = (const float*)d_in[2];
    const float* W  = (const float*)d_in[3];
    float* out = (float*)d_out;
    float* ws  = (float*)d_ws;

    float* partials = ws;            // NPART floats
    float* wsum     = ws + NPART;    // 1 float
    float* ssums    = ws + NPART + 1;// 3*768 floats (s1|s2|s3)

    const int nW = in_sizes[3];      // 256^3

    hipLaunchKernelGGL(reduce_w_partials, dim3(NPART), dim3(256), 0, stream,
                       W, partials, nW);
    hipLaunchKernelGGL(reduce_w_final, dim3(1), dim3(256), 0, stream,
                       partials, wsum, NPART);
    hipLaunchKernelGGL(row_sums, dim3((3 * NROWS * 32) / 256), dim3(256), 0, stream,
                       m1, m2, m3, ssums);
    hipLaunchKernelGGL(trilinear_write, dim3(Bsz * NT * NT), dim3(256), 0, stream,
                       ssums, wsum, out);
}